// SparseGraphSAGE_86457691668756
// MI455X (gfx1250) — compile-verified
//
#include <hip/hip_runtime.h>
#include <hip/hip_bf16.h>

// ---------------------------------------------------------------------------
// SparseGraphSAGE for MI455X (gfx1250, wave32, WMMA)
//   deg/deg_inv  : atomic segment-sum over edges
//   SpMM         : wave-per-edge float4 gather + global_atomic_add_f32 scatter
//   dense GEMMs  : V_WMMA_F32_16X16X4_F32 (fp32 in / fp32 acc — matches ref)
//                  with pre-transposed weights so each B-fragment is ONE b64 load
//   decoder      : WMMA (K=256) + VALU dot with Wd2
// ---------------------------------------------------------------------------

typedef __attribute__((ext_vector_type(2))) float v2f;
typedef __attribute__((ext_vector_type(8))) float v8f;

#define N_NODES 50000
#define N_EDGES 800000
#define HID     128
#define N_POS   100000

// ---------------- utility: zero a float region (grid-stride) ---------------
__global__ void zero_kernel(float* __restrict__ p, long long n) {
  long long i = (long long)blockIdx.x * blockDim.x + threadIdx.x;
  long long stride = (long long)gridDim.x * blockDim.x;
  for (; i < n; i += stride) p[i] = 0.0f;
}

// ---------------- transpose KxN (N=128) -> Wt[n*K + k] ----------------------
__global__ void transpose_kernel(const float* __restrict__ W, float* __restrict__ Wt,
                                 int K) {
  int id = blockIdx.x * blockDim.x + threadIdx.x;   // output-linear: coalesced write
  if (id < K * HID) {
    int n = id / K;
    int k = id - n * K;
    Wt[id] = W[k * HID + n];
  }
}

// ---------------- degree accumulation: deg[row] += val ----------------------
__global__ void deg_kernel(const int* __restrict__ idx, const float* __restrict__ vals,
                           float* __restrict__ deg, int nedges) {
  int e = blockIdx.x * blockDim.x + threadIdx.x;
  if (e < nedges) atomicAdd(&deg[idx[2 * e]], vals[e]);
}

// ---------------- deg -> 1/max(deg,1) in place ------------------------------
__global__ void deginv_kernel(float* __restrict__ deg, int n) {
  int i = blockIdx.x * blockDim.x + threadIdx.x;
  if (i < n) deg[i] = 1.0f / fmaxf(deg[i], 1.0f);
}

// ---------------- SpMM: out[row] += val * x[col]  (one wave per edge) ------
__global__ void spmm_kernel(const int* __restrict__ idx, const float* __restrict__ vals,
                            const float* __restrict__ x, float* __restrict__ out,
                            int nedges) {
  int gw   = (blockIdx.x * blockDim.x + threadIdx.x) >> 5;  // edge id
  int lane = threadIdx.x & 31;
  if (gw >= nedges) return;
  int   row = idx[2 * gw];
  int   col = idx[2 * gw + 1];
  float v   = vals[gw];
  int   f   = lane << 2;                                    // 4 feats per lane
  const float4 xv = *(const float4*)(x + (long long)col * HID + f);
  float* o = out + (long long)row * HID + f;
  atomicAdd(o + 0, v * xv.x);
  atomicAdd(o + 1, v * xv.y);
  atomicAdd(o + 2, v * xv.z);
  atomicAdd(o + 3, v * xv.w);
}

// ---------------- Y = [relu]((X .* scale_row) @ W + b)  via WMMA f32 -------
// Wt is the TRANSPOSED weight: Wt[n*128 + k] == W[k][n].
// grid.x = M/16 row tiles; block = 256 threads = 8 wave32, wave w -> cols w*16..
__global__ void gemm_wmma_kernel(const float* __restrict__ X, const float* __restrict__ scale,
                                 const float* __restrict__ Wt, const float* __restrict__ bias,
                                 float* __restrict__ Y, int do_relu) {
  __shared__ float Xs[16][132];               // pad 128->132: conflict-free col reads
  const int tile_m = blockIdx.x;
  const int t      = threadIdx.x;

  // cooperative load of the 16x128 A-strip, pre-scaled by deg_inv
  {
    int   r    = t >> 4;                      // 0..15
    int   cseg = (t & 15) << 3;               // 8 floats each
    float s    = scale[tile_m * 16 + r];
    const float* src = X + ((long long)tile_m * 16 + r) * HID + cseg;
    float4 v0 = *(const float4*)(src);
    float4 v1 = *(const float4*)(src + 4);
    float* dst = &Xs[r][cseg];
    dst[0] = v0.x * s; dst[1] = v0.y * s; dst[2] = v0.z * s; dst[3] = v0.w * s;
    dst[4] = v1.x * s; dst[5] = v1.y * s; dst[6] = v1.z * s; dst[7] = v1.w * s;
  }
  __syncthreads();

  const int lane = t & 31;
  const int wid  = t >> 5;
  const int m    = lane & 15;
  const int kh   = (lane >> 4) << 1;          // 0 or 2 (K half-select)
  const int n    = (wid << 4) + (lane & 15);  // output column

  const float* wrow = Wt + n * HID;           // this lane's weight column (contiguous)

  v8f c = {};
#pragma unroll
  for (int k0 = 0; k0 < HID; k0 += 4) {
    int k = k0 + kh;
    v2f a; a.x = Xs[m][k]; a.y = Xs[m][k + 1];
    v2f b = *(const v2f*)(wrow + k);          // single global_load_b64
    c = __builtin_amdgcn_wmma_f32_16x16x4_f32(false, a, false, b, (short)0, c,
                                              false, false);
  }

  const float bn      = bias[n];
  const int   rowbase = tile_m * 16 + ((lane >> 4) << 3);
#pragma unroll
  for (int i = 0; i < 8; ++i) {
    float v = c[i] + bn;
    if (do_relu) v = fmaxf(v, 0.0f);
    Y[(long long)(rowbase + i) * HID + n] = v;
  }
}

// ---------------- decoder: out[p] = relu([h2[u],h2[v]] @ Wd1 + bd1) @ Wd2 + bd2
// Wd1t[n*256 + k] == Wd1[k][n]. block = 256 thr = 8 waves, 16 pairs; K=256, N=128
__global__ void decode_kernel(const float* __restrict__ h2,
                              const int* __restrict__ U, const int* __restrict__ V,
                              const float* __restrict__ Wd1t, const float* __restrict__ bd1,
                              const float* __restrict__ Wd2, const float* __restrict__ bd2,
                              float* __restrict__ out) {
  __shared__ float Zs[16][260];               // 16 x 256 (padded)
  __shared__ float Hs[16][HID];               // relu hidden
  __shared__ float red[16][16];

  const int base = blockIdx.x * 16;
  const int t    = threadIdx.x;

  // gather z = [h2[u], h2[v]] : thread t -> pair i, 16-float segment seg
  {
    int i   = t >> 4;
    int seg = t & 15;
    const float* src = (seg < 8)
        ? h2 + (long long)U[base + i] * HID + (seg << 4)
        : h2 + (long long)V[base + i] * HID + ((seg - 8) << 4);
    float* dst = &Zs[i][seg << 4];
#pragma unroll
    for (int j = 0; j < 16; j += 4) {
      float4 v = *(const float4*)(src + j);
      dst[j] = v.x; dst[j + 1] = v.y; dst[j + 2] = v.z; dst[j + 3] = v.w;
    }
  }
  __syncthreads();

  const int lane = t & 31;
  const int wid  = t >> 5;
  const int m    = lane & 15;
  const int kh   = (lane >> 4) << 1;
  const int n    = (wid << 4) + (lane & 15);

  const float* wrow = Wd1t + n * (2 * HID);   // contiguous 256-float column of Wd1

  v8f c = {};
#pragma unroll
  for (int k0 = 0; k0 < 2 * HID; k0 += 4) {
    int k = k0 + kh;
    v2f a; a.x = Zs[m][k]; a.y = Zs[m][k + 1];
    v2f b = *(const v2f*)(wrow + k);          // single global_load_b64
    c = __builtin_amdgcn_wmma_f32_16x16x4_f32(false, a, false, b, (short)0, c,
                                              false, false);
  }

  const float bn    = bd1[n];
  const int   rbase = (lane >> 4) << 3;
#pragma unroll
  for (int i = 0; i < 8; ++i) Hs[rbase + i][n] = fmaxf(c[i] + bn, 0.0f);
  __syncthreads();

  // hidden @ Wd2 : 16 partial dots per pair, then reduce
  {
    int i   = t >> 4;
    int seg = t & 15;
    float p = 0.0f;
#pragma unroll
    for (int j = 0; j < 8; ++j) p += Hs[i][(seg << 3) + j] * Wd2[(seg << 3) + j];
    red[i][seg] = p;
  }
  __syncthreads();
  if (t < 16) {
    float s = bd2[0];
#pragma unroll
    for (int j = 0; j < 16; ++j) s += red[t][j];
    out[base + t] = s;
  }
}

// ---------------------------------------------------------------------------
extern "C" void kernel_launch(void* const* d_in, const int* in_sizes, int n_in,
                              void* d_out, int out_size, void* d_ws, size_t ws_size,
                              hipStream_t stream) {
  const int*   adj_idx = (const int*)  d_in[0];   // (800000, 2) int32
  const float* adj_val = (const float*)d_in[1];   // (800000,)
  const float* h       = (const float*)d_in[3];   // (50000, 128)
  const int*   pos_u   = (const int*)  d_in[4];
  const int*   pos_v   = (const int*)  d_in[5];
  const int*   neg_u   = (const int*)  d_in[6];
  const int*   neg_v   = (const int*)  d_in[7];
  const float* W1      = (const float*)d_in[8];
  const float* b1      = (const float*)d_in[9];
  const float* W2      = (const float*)d_in[10];
  const float* b2      = (const float*)d_in[11];
  const float* Wd1     = (const float*)d_in[12];
  const float* bd1     = (const float*)d_in[13];
  const float* Wd2     = (const float*)d_in[14];
  const float* bd2     = (const float*)d_in[15];

  float* outp = (float*)d_out;
  float* pos_scores = outp;                        // [0, 100000)
  float* neg_scores = outp + N_POS;                // [100000, 200000)
  float* h2         = outp + 2 * N_POS;            // [200000, 200000 + 6.4M)

  // workspace layout (floats)
  float* ws      = (float*)d_ws;
  float* deg_inv = ws;                             // 50176 (padded)
  float* tmp     = ws + 50176;                     // 6,400,000  spmm accumulator
  float* h1      = tmp + (long long)N_NODES * HID; // 6,400,000
  float* W1t     = h1 + (long long)N_NODES * HID;  // 16384
  float* W2t     = W1t + HID * HID;                // 16384
  float* Wd1t    = W2t + HID * HID;                // 32768

  const long long feat_elems = (long long)N_NODES * HID;

  // 0) transpose the three GEMM weights (one-time, tiny)
  transpose_kernel<<<(HID * HID + 255) / 256, 256, 0, stream>>>(W1, W1t, HID);
  transpose_kernel<<<(HID * HID + 255) / 256, 256, 0, stream>>>(W2, W2t, HID);
  transpose_kernel<<<(2 * HID * HID + 255) / 256, 256, 0, stream>>>(Wd1, Wd1t, 2 * HID);
  // 1) zero deg + tmp
  zero_kernel<<<2048, 256, 0, stream>>>(ws, 50176 + feat_elems);
  // 2) deg accumulation + invert
  deg_kernel<<<(N_EDGES + 255) / 256, 256, 0, stream>>>(adj_idx, adj_val, deg_inv, N_EDGES);
  deginv_kernel<<<(N_NODES + 255) / 256, 256, 0, stream>>>(deg_inv, N_NODES);
  // 3) tmp = SpMM(h)
  spmm_kernel<<<N_EDGES / 8, 256, 0, stream>>>(adj_idx, adj_val, h, tmp, N_EDGES);
  // 4) h1 = relu((tmp .* deg_inv) @ W1 + b1)
  gemm_wmma_kernel<<<N_NODES / 16, 256, 0, stream>>>(tmp, deg_inv, W1t, b1, h1, 1);
  // 5) zero tmp, tmp = SpMM(h1)
  zero_kernel<<<2048, 256, 0, stream>>>(tmp, feat_elems);
  spmm_kernel<<<N_EDGES / 8, 256, 0, stream>>>(adj_idx, adj_val, h1, tmp, N_EDGES);
  // 6) h2 = (tmp .* deg_inv) @ W2 + b2   (written straight into d_out)
  gemm_wmma_kernel<<<N_NODES / 16, 256, 0, stream>>>(tmp, deg_inv, W2t, b2, h2, 0);
  // 7) decode positive + negative pairs
  decode_kernel<<<N_POS / 16, 256, 0, stream>>>(h2, pos_u, pos_v, Wd1t, bd1, Wd2, bd2, pos_scores);
  decode_kernel<<<N_POS / 16, 256, 0, stream>>>(h2, neg_u, neg_v, Wd1t, bd1, Wd2, bd2, neg_scores);
}